// adapt_layer_off_18966575579364
// MI455X (gfx1250) — compile-verified
//
#include <hip/hip_runtime.h>
#include <hip/hip_bf16.h>
#include <cstdint>

// ---------------------------------------------------------------------------
// adapt_layer_off for MI455X (gfx1250, wave32)
// B=16, C=64, N=16384, NUM_NODE=S=64, NSAMPLE=K=64, K_UP=3
// Dominant GEMM (y = W x + b over 262144 points) done with V_WMMA_F32_16X16X4_F32.
// Workspace need: ~66 MiB (y buffer 64MiB + small index/stat arrays).
// ---------------------------------------------------------------------------

#define Bq 16
#define Cch 64
#define Nn 16384
#define Sn 64
#define Kn 64
#define R2 0.09f

typedef __attribute__((ext_vector_type(2))) float v2f;
typedef __attribute__((ext_vector_type(8))) float v8f;

#if defined(__gfx1250__) && __has_builtin(__builtin_amdgcn_wmma_f32_16x16x4_f32)
#define HAVE_WMMA_F32X4 1
#else
#define HAVE_WMMA_F32X4 0
#endif

// ---------------- 1. farthest point sampling (1 block per batch) -----------
__global__ __launch_bounds__(256) void fps_kernel(const float* __restrict__ loc,
                                                  float* __restrict__ dist,
                                                  int* __restrict__ fidx) {
  __shared__ float sval[256];
  __shared__ int sidx[256];
  __shared__ int sfar;
  const int b = blockIdx.x, tid = threadIdx.x;
  const float* lb = loc + (size_t)b * 3 * Nn;
  float* db = dist + (size_t)b * Nn;
  for (int n = tid; n < Nn; n += 256) db[n] = 1e10f;
  if (tid == 0) sfar = 0;
  __syncthreads();
  for (int it = 0; it < Sn; ++it) {
    const int far = sfar;
    if (tid == 0) fidx[b * Sn + it] = far;
    const float cx = lb[far], cy = lb[Nn + far], cz = lb[2 * Nn + far];
    float bv = -1.0f; int bi = 0;
    for (int n = tid; n < Nn; n += 256) {
      const float dx = lb[n] - cx, dy = lb[Nn + n] - cy, dz = lb[2 * Nn + n] - cz;
      const float nd = fminf(db[n], dx * dx + dy * dy + dz * dz);
      db[n] = nd;
      if (nd > bv) { bv = nd; bi = n; }   // strided ascending -> first-max kept
    }
    sval[tid] = bv; sidx[tid] = bi;
    __syncthreads();
    for (int st = 128; st > 0; st >>= 1) {
      if (tid < st) {
        if (sval[tid + st] > sval[tid] ||
            (sval[tid + st] == sval[tid] && sidx[tid + st] < sidx[tid])) {
          sval[tid] = sval[tid + st]; sidx[tid] = sidx[tid + st];
        }
      }
      __syncthreads();
    }
    if (tid == 0) sfar = sidx[0];
    __syncthreads();
  }
}

// ---------------- 2. gather sampled loc/fea --------------------------------
__global__ __launch_bounds__(64) void gather_fps_kernel(const float* __restrict__ loc,
                                                        const float* __restrict__ fea,
                                                        const int* __restrict__ fidx,
                                                        float* __restrict__ fpl,
                                                        float* __restrict__ fpf) {
  const int b = blockIdx.x >> 6, s = blockIdx.x & 63, c = threadIdx.x;
  const int f = fidx[b * Sn + s];
  fpf[((size_t)b * Cch + c) * Sn + s] = fea[((size_t)b * Cch + c) * Nn + f];
  if (c < 3) fpl[((size_t)b * 3 + c) * Sn + s] = loc[((size_t)b * 3 + c) * Nn + f];
}

// ---------------- 3. ball query (thread per center) ------------------------
__global__ __launch_bounds__(256) void qball_kernel(const float* __restrict__ loc,
                                                    const float* __restrict__ fpl,
                                                    int* __restrict__ gidx) {
  const int t = blockIdx.x * 256 + threadIdx.x;
  if (t >= Bq * Sn) return;
  const int b = t >> 6, s = t & 63;
  const float* lb = loc + (size_t)b * 3 * Nn;
  const float cx = fpl[((size_t)b * 3 + 0) * Sn + s];
  const float cy = fpl[((size_t)b * 3 + 1) * Sn + s];
  const float cz = fpl[((size_t)b * 3 + 2) * Sn + s];
  int* g = gidx + ((size_t)b * Sn + s) * Kn;
  int found = 0;
  for (int n = 0; n < Nn && found < Kn; ++n) {
    const float dx = lb[n] - cx, dy = lb[Nn + n] - cy, dz = lb[2 * Nn + n] - cz;
    if (dx * dx + dy * dy + dz * dz <= R2) g[found++] = n;
  }
  const int first = (found > 0) ? g[0] : 0;
  for (int k = found; k < Kn; ++k) g[k] = first;
}

// ---------------- 4. semantic offset + node_loc (block per center) ---------
__global__ __launch_bounds__(64) void offset_kernel(const float* __restrict__ fea,
                                                    const float* __restrict__ loc,
                                                    const float* __restrict__ fpf,
                                                    const float* __restrict__ fpl,
                                                    const int* __restrict__ gidx,
                                                    const float* __restrict__ woff,
                                                    float* __restrict__ node_loc,
                                                    float* __restrict__ out_off) {
  __shared__ float sff[Cch];
  __shared__ float sw[3][Cch];
  __shared__ float red[3][64];
  const int b = blockIdx.x >> 6, s = blockIdx.x & 63, k = threadIdx.x;
  sff[k] = fpf[((size_t)b * Cch + k) * Sn + s];
  for (int o = 0; o < 3; ++o) sw[o][k] = woff[o * Cch + k];
  __syncthreads();
  const int idx = gidx[((size_t)b * Sn + s) * Kn + k];
  const float* fb = fea + (size_t)b * Cch * Nn;
  float t0 = 0.f, t1 = 0.f, t2 = 0.f;
  for (int c = 0; c < Cch; ++c) {
    const float gf = fb[(size_t)c * Nn + idx] - sff[c];
    t0 += sw[0][c] * gf; t1 += sw[1][c] * gf; t2 += sw[2][c] * gf;
  }
  t0 = tanhf(t0); t1 = tanhf(t1); t2 = tanhf(t2);
  const float* lb = loc + (size_t)b * 3 * Nn;
  red[0][k] = t0 * (lb[idx] - fpl[((size_t)b * 3 + 0) * Sn + s]);
  red[1][k] = t1 * (lb[Nn + idx] - fpl[((size_t)b * 3 + 1) * Sn + s]);
  red[2][k] = t2 * (lb[2 * Nn + idx] - fpl[((size_t)b * 3 + 2) * Sn + s]);
  __syncthreads();
  for (int st = 32; st > 0; st >>= 1) {
    if (k < st)
      for (int o = 0; o < 3; ++o) red[o][k] += red[o][k + st];
    __syncthreads();
  }
  if (k < 3) {
    const float off = red[k][0] * (1.0f / 64.0f);
    out_off[((size_t)b * 3 + k) * Sn + s] = off;
    node_loc[((size_t)b * 3 + k) * Sn + s] = fpl[((size_t)b * 3 + k) * Sn + s] + off;
  }
}

// ---------------- 5. 64-NN selection (block per center, dyn LDS dists) -----
__global__ __launch_bounds__(256) void knn_kernel(const float* __restrict__ loc,
                                                  const float* __restrict__ node_loc,
                                                  int* __restrict__ gidx2) {
  extern __shared__ float sdist[];   // Nn floats
  __shared__ float rv[256];
  __shared__ int ri[256];
  const int b = blockIdx.x >> 6, s = blockIdx.x & 63, tid = threadIdx.x;
  const float* lb = loc + (size_t)b * 3 * Nn;
  const float cx = node_loc[((size_t)b * 3 + 0) * Sn + s];
  const float cy = node_loc[((size_t)b * 3 + 1) * Sn + s];
  const float cz = node_loc[((size_t)b * 3 + 2) * Sn + s];
  for (int n = tid; n < Nn; n += 256) {
    const float dx = lb[n] - cx, dy = lb[Nn + n] - cy, dz = lb[2 * Nn + n] - cz;
    sdist[n] = dx * dx + dy * dy + dz * dz;
  }
  __syncthreads();
  int* g = gidx2 + ((size_t)b * Sn + s) * Kn;
  for (int sel = 0; sel < Kn; ++sel) {
    float bv = 3.0e38f; int bi = 0x7fffffff;
    for (int n = tid; n < Nn; n += 256) {
      const float d = sdist[n];
      if (d < bv || (d == bv && n < bi)) { bv = d; bi = n; }
    }
    rv[tid] = bv; ri[tid] = bi;
    __syncthreads();
    for (int st = 128; st > 0; st >>= 1) {
      if (tid < st) {
        if (rv[tid + st] < rv[tid] ||
            (rv[tid + st] == rv[tid] && ri[tid + st] < ri[tid])) {
          rv[tid] = rv[tid + st]; ri[tid] = ri[tid + st];
        }
      }
      __syncthreads();
    }
    if (tid == 0) { g[sel] = ri[0]; sdist[ri[0]] = 1.0e30f; }
    __syncthreads();
  }
}

// ---------------- 6. zero BN accumulators ----------------------------------
__global__ void zero_kernel(float* __restrict__ p) { p[threadIdx.x] = 0.0f; }

// ---------------- 7. residual GEMM via WMMA f32 16x16x4 + BN partials ------
__global__ __launch_bounds__(128) void gemm_bn_kernel(const float* __restrict__ fea,
                                                      const float* __restrict__ w,
                                                      const float* __restrict__ bias,
                                                      float* __restrict__ y,
                                                      float* __restrict__ sums) {
  const int wave = threadIdx.x >> 5;           // 0..3 -> o tile
  const int lane = threadIdx.x & 31;
  const int b = blockIdx.x >> 10;
  const int n_tile = (blockIdx.x & 1023) << 4; // 1024 n-tiles of 16
  const int o_tile = wave << 4;
  const int half = lane >> 4;                  // K-pair select
  const int l15 = lane & 15;

  v8f acc = {};
  const float* wrow = w + (size_t)(o_tile + l15) * Cch + half * 2;
  const float* fcol = fea + (size_t)b * Cch * Nn + (size_t)(half * 2) * Nn + n_tile + l15;
#if HAVE_WMMA_F32X4
  #pragma unroll
  for (int k = 0; k < 16; ++k) {
    const int c = k * 4;
    v2f a;  a.x = wrow[c];               a.y = wrow[c + 1];
    v2f bb; bb.x = fcol[(size_t)c * Nn]; bb.y = fcol[(size_t)(c + 1) * Nn];
    acc = __builtin_amdgcn_wmma_f32_16x16x4_f32(false, a, false, bb,
                                                (short)0, acc, false, false);
  }
#else
  // scalar fallback (host pass / missing builtin)
  for (int r = 0; r < 8; ++r) {
    const int o = o_tile + r + half * 8;
    float a = 0.f;
    for (int c = 0; c < Cch; ++c)
      a += w[(size_t)o * Cch + c] * fea[((size_t)b * Cch + c) * Nn + n_tile + l15];
    acc[r] = a;
  }
#endif
  float* yb = y + (size_t)b * Cch * Nn;
  #pragma unroll
  for (int r = 0; r < 8; ++r) {
    const int o = o_tile + r + half * 8;
    const int n = n_tile + l15;
    const float v = acc[r] + bias[o];
    yb[(size_t)o * Nn + n] = v;
    float s = v, ss = v * v;
    for (int off = 1; off < 16; off <<= 1) {      // reduce 16-lane row group
      s += __shfl_xor(s, off, 32);
      ss += __shfl_xor(ss, off, 32);
    }
    if (l15 == 0) {
      atomicAdd(&sums[o], s);
      atomicAdd(&sums[64 + o], ss);
    }
  }
}

// ---------------- 8. BN stats -> per-channel scale/shift -------------------
__global__ __launch_bounds__(64) void bn_stats_kernel(const float* __restrict__ sums,
                                                      const float* __restrict__ g,
                                                      const float* __restrict__ be,
                                                      float* __restrict__ ac) {
  const int o = threadIdx.x;
  const float cnt = (float)(Bq * Nn);
  const float mean = sums[o] / cnt;
  const float var = sums[64 + o] / cnt - mean * mean;
  const float inv = rsqrtf(var + 1e-5f);
  const float a = g[o] * inv;
  ac[o] = a;
  ac[64 + o] = be[o] - mean * a;
}

// ---------------- 9. normalize + relu (in place) ---------------------------
__global__ __launch_bounds__(256) void bn_apply_kernel(float* __restrict__ y,
                                                       const float* __restrict__ ac) {
  const size_t i = (size_t)blockIdx.x * 256 + threadIdx.x;
  const int o = (int)((i >> 14) & 63);
  y[i] = fmaxf(ac[o] * y[i] + ac[64 + o], 0.0f);
}

// ---------------- 10. gather + max -> node_fea -----------------------------
__global__ __launch_bounds__(64) void nodefea_kernel(const float* __restrict__ res,
                                                     const int* __restrict__ gidx2,
                                                     float* __restrict__ node_fea) {
  const int b = blockIdx.x >> 6, s = blockIdx.x & 63, o = threadIdx.x;
  const int* g = gidx2 + ((size_t)b * Sn + s) * Kn;
  const float* rb = res + ((size_t)b * Cch + o) * Nn;
  float m = -3.0e38f;
  for (int k = 0; k < Kn; ++k) m = fmaxf(m, rb[g[k]]);
  node_fea[((size_t)b * Cch + o) * Sn + s] = m;
}

// ---------------- 11. 3-NN inverse-distance upsample + concat --------------
__global__ __launch_bounds__(256) void upsample_kernel(const float* __restrict__ loc,
                                                       const float* __restrict__ fea,
                                                       const float* __restrict__ node_loc,
                                                       const float* __restrict__ node_fea,
                                                       float* __restrict__ out) {
  __shared__ float nl[3][Sn];
  __shared__ float nf[Cch][Sn];   // 16 KB
  const int b = blockIdx.x >> 6, chunk = blockIdx.x & 63, tid = threadIdx.x;
  for (int i = tid; i < 3 * Sn; i += 256)
    nl[i / Sn][i % Sn] = node_loc[(size_t)b * 3 * Sn + i];
  for (int i = tid; i < Cch * Sn; i += 256)
    nf[i >> 6][i & 63] = node_fea[(size_t)b * Cch * Sn + i];
  __syncthreads();
  const int n = chunk * 256 + tid;
  const float* lb = loc + (size_t)b * 3 * Nn;
  const float px = lb[n], py = lb[Nn + n], pz = lb[2 * Nn + n];
  float d0 = 3.0e38f, d1 = 3.0e38f, d2 = 3.0e38f;
  int i0 = 0, i1 = 0, i2 = 0;
  for (int j = 0; j < Sn; ++j) {
    const float dx = px - nl[0][j], dy = py - nl[1][j], dz = pz - nl[2][j];
    const float d = dx * dx + dy * dy + dz * dz;
    if (d < d0)      { d2 = d1; i2 = i1; d1 = d0; i1 = i0; d0 = d; i0 = j; }
    else if (d < d1) { d2 = d1; i2 = i1; d1 = d;  i1 = j; }
    else if (d < d2) { d2 = d;  i2 = j; }
  }
  float w0 = 1.0f / fmaxf(d0, 1e-10f);
  float w1 = 1.0f / fmaxf(d1, 1e-10f);
  float w2 = 1.0f / fmaxf(d2, 1e-10f);
  const float wsum = w0 + w1 + w2;
  w0 /= wsum; w1 /= wsum; w2 /= wsum;
  const float* fb = fea + (size_t)b * Cch * Nn;
  float* ob = out + (size_t)b * 128 * Nn;
  for (int c = 0; c < Cch; ++c) {
    ob[(size_t)c * Nn + n] = fb[(size_t)c * Nn + n];
    ob[(size_t)(Cch + c) * Nn + n] = w0 * nf[c][i0] + w1 * nf[c][i1] + w2 * nf[c][i2];
  }
}

// ---------------------------------------------------------------------------
extern "C" void kernel_launch(void* const* d_in, const int* in_sizes, int n_in,
                              void* d_out, int out_size, void* d_ws, size_t ws_size,
                              hipStream_t stream) {
  const float* input_fea = (const float*)d_in[0];   // (16,64,16384)
  const float* input_loc = (const float*)d_in[1];   // (16,3,16384)
  const float* w_res     = (const float*)d_in[2];   // (64,64)
  const float* b_res     = (const float*)d_in[3];   // (64)
  const float* g_res     = (const float*)d_in[4];   // (64)
  const float* be_res    = (const float*)d_in[5];   // (64)
  const float* w_off     = (const float*)d_in[6];   // (3,64)

  float* out = (float*)d_out;
  float* out_fea  = out;                                          // (16,128,16384,1)
  float* out_nf   = out + (size_t)Bq * 128 * Nn;                  // (16,64,64,1)
  float* out_noff = out_nf + (size_t)Bq * Cch * Sn;               // (16,3,64)

  char* ws = (char*)d_ws;
  float* y       = (float*)(ws);                                   // 64 MiB
  float* dist    = (float*)(ws + 67108864);                        // 1 MiB
  int*   fidx    = (int*)  (ws + 68157440);                        // 4 KiB
  float* fpl     = (float*)(ws + 68161536);                        // 12 KiB
  float* fpf     = (float*)(ws + 68173824);                        // 256 KiB
  int*   gidx    = (int*)  (ws + 68435968);                        // 256 KiB
  int*   gidx2   = (int*)  (ws + 68698112);                        // 256 KiB
  float* nodeloc = (float*)(ws + 68960256);                        // 12 KiB
  float* sums    = (float*)(ws + 68972544);                        // 128 floats
  float* bn_ac   = (float*)(ws + 68973056);                        // 128 floats

  fps_kernel<<<Bq, 256, 0, stream>>>(input_loc, dist, fidx);
  gather_fps_kernel<<<Bq * Sn, 64, 0, stream>>>(input_loc, input_fea, fidx, fpl, fpf);
  qball_kernel<<<(Bq * Sn + 255) / 256, 256, 0, stream>>>(input_loc, fpl, gidx);
  offset_kernel<<<Bq * Sn, 64, 0, stream>>>(input_fea, input_loc, fpf, fpl, gidx,
                                            w_off, nodeloc, out_noff);
  knn_kernel<<<Bq * Sn, 256, Nn * sizeof(float), stream>>>(input_loc, nodeloc, gidx2);
  zero_kernel<<<1, 128, 0, stream>>>(sums);
  gemm_bn_kernel<<<Bq * (Nn / 16), 128, 0, stream>>>(input_fea, w_res, b_res, y, sums);
  bn_stats_kernel<<<1, 64, 0, stream>>>(sums, g_res, be_res, bn_ac);
  bn_apply_kernel<<<(Bq * Cch * Nn) / 256, 256, 0, stream>>>(y, bn_ac);
  nodefea_kernel<<<Bq * Sn, 64, 0, stream>>>(y, gidx2, out_nf);
  upsample_kernel<<<Bq * (Nn / 256), 256, 0, stream>>>(input_loc, input_fea,
                                                       nodeloc, out_nf, out_fea);
}